// Server_encoder_multi_25752623907302
// MI455X (gfx1250) — compile-verified
//
#include <hip/hip_runtime.h>

#define DIM 128
#define LDS_STRIDE 132   // 132 % 64 == 4 -> conflict-free b64 LDS reads of W tiles

typedef __attribute__((ext_vector_type(2))) float v2f;
typedef __attribute__((ext_vector_type(8))) float v8f;

__device__ __forceinline__ void atomic_add_f32(float* p, float v) {
  // hardware global_atomic_add_f32 (no CAS loop)
  unsafeAtomicAdd(p, v);
}

// ---------------------------------------------------------------- copy (agg = x)
__global__ __launch_bounds__(256) void copy_f4_kernel(const float4* __restrict__ src,
                                                      float4* __restrict__ dst, int n4) {
  int i = blockIdx.x * blockDim.x + threadIdx.x;
  if (i < n4) dst[i] = src[i];
}

// ---------------------------------------------------------------- edge scatter-add
// one wave per edge, each lane handles 4 channels (float4 gather + 4 f32 atomics)
__global__ __launch_bounds__(256) void scatter_add_kernel(const float* __restrict__ x,
                                                          const int* __restrict__ srcI,
                                                          const int* __restrict__ dstI,
                                                          float* __restrict__ agg,
                                                          int n_edges) {
  int tid = blockIdx.x * blockDim.x + threadIdx.x;
  int e = tid >> 5;
  if (e >= n_edges) return;
  int q = (tid & 31) * 4;                       // channel quad
  int s = srcI[e];                              // wave-uniform -> scalar load
  int d = dstI[e];
  const float4 v = *(const float4*)(x + (size_t)s * DIM + q);
  float* out = agg + (size_t)d * DIM + q;
  atomic_add_f32(out + 0, v.x);
  atomic_add_f32(out + 1, v.y);
  atomic_add_f32(out + 2, v.z);
  atomic_add_f32(out + 3, v.w);
}

// ---------------------------------------------------------------- WMMA f32 GEMM
// out[n][j] = act( sum_k in[n][k] * W[j][k] + bias[j] )
// wave-level: 16 rows (M tile) x 128 cols via 8 N-tiles, K=128 as 32 x WMMA_F32_16X16X4_F32
template <bool LRELU>
__global__ __launch_bounds__(256) void gemm_kernel(const float* __restrict__ A,
                                                   const float* __restrict__ W,
                                                   const float* __restrict__ bias,
                                                   float* __restrict__ out,
                                                   int n_rows) {
  extern __shared__ float ldsW[];  // [128][LDS_STRIDE]
  // cooperative stage of W (row-major) into padded LDS
  for (int i = threadIdx.x; i < DIM * DIM; i += 256) {
    int r = i >> 7, c = i & 127;
    ldsW[r * LDS_STRIDE + c] = W[i];
  }
  __syncthreads();

  const int wave = threadIdx.x >> 5;
  const int lane = threadIdx.x & 31;
  const int m0 = (blockIdx.x * 8 + wave) * 16;
  if (m0 >= n_rows) return;                 // wave-uniform; EXEC stays all-1s for WMMA

  const int lr = lane & 15;                 // A: row in tile / B: col in tile
  const int hi = lane >> 4;                 // K half selector (0 -> K+0/1, 1 -> K+2/3)

  v8f acc[8] = {};
  const float* arow  = A + (size_t)(m0 + lr) * DIM + hi * 2;
  const float* wbase = ldsW + (size_t)lr * LDS_STRIDE + hi * 2;

  for (int k = 0; k < 32; ++k) {            // K steps of 4
    v2f a;
    a.x = arow[0];
    a.y = arow[1];
    arow += 4;
#pragma unroll
    for (int j = 0; j < 8; ++j) {           // N tiles
      const float* wp = wbase + (j * 16) * LDS_STRIDE + k * 4;
      v2f b;
      b.x = wp[0];
      b.y = wp[1];
      acc[j] = __builtin_amdgcn_wmma_f32_16x16x4_f32(
          /*neg_a=*/false, a, /*neg_b=*/false, b,
          /*c_mod=*/(short)0, acc[j], /*reuse_a=*/false, /*reuse_b=*/false);
    }
  }

  // epilogue: bias + optional LeakyReLU, C layout: VGPR r -> (row m0+r+8*hi, col 16j+lr)
#pragma unroll
  for (int j = 0; j < 8; ++j) {
    const float bv = bias[j * 16 + lr];
#pragma unroll
    for (int r = 0; r < 8; ++r) {
      float v = acc[j][r] + bv;
      if (LRELU) v = (v > 0.0f) ? v : v * 0.01f;
      out[(size_t)(m0 + r + 8 * hi) * DIM + (j * 16 + lr)] = v;
    }
  }
}

// ---------------------------------------------------------------- BatchNorm pieces
__global__ void bn_zero_kernel(float* stats) {
  if (threadIdx.x < 256) stats[threadIdx.x] = 0.0f;
}

__global__ __launch_bounds__(128) void bn_stats_kernel(const float* __restrict__ h,
                                                       float* __restrict__ stats, int n) {
  int c = threadIdx.x;  // 0..127, coalesced across channels
  float s = 0.0f, s2 = 0.0f;
  for (int r = blockIdx.x; r < n; r += gridDim.x) {
    float v = h[(size_t)r * DIM + c];
    s += v;
    s2 += v * v;
  }
  atomic_add_f32(&stats[c], s);
  atomic_add_f32(&stats[DIM + c], s2);
}

__global__ void bn_finalize_kernel(const float* __restrict__ stats,
                                   const float* __restrict__ gamma,
                                   const float* __restrict__ beta,
                                   float* __restrict__ ss, float inv_n) {
  int c = threadIdx.x;
  if (c < DIM) {
    float mean = stats[c] * inv_n;
    float var = stats[DIM + c] * inv_n - mean * mean;    // biased variance
    float sc = gamma[c] * rsqrtf(var + 1e-4f);
    ss[c] = sc;
    ss[DIM + c] = beta[c] - mean * sc;
  }
}

__global__ __launch_bounds__(256) void bn_apply_kernel(float* __restrict__ h,
                                                       const float* __restrict__ ss,
                                                       int total) {
  int i = blockIdx.x * blockDim.x + threadIdx.x;
  if (i < total) {
    int c = i & 127;
    h[i] = h[i] * ss[c] + ss[DIM + c];
  }
}

// ---------------------------------------------------------------- host orchestration
extern "C" void kernel_launch(void* const* d_in, const int* in_sizes, int n_in,
                              void* d_out, int out_size, void* d_ws, size_t ws_size,
                              hipStream_t stream) {
  const float* x     = (const float*)d_in[0];
  const int*   edge  = (const int*)d_in[1];
  const float* W1    = (const float*)d_in[2];
  const float* b1    = (const float*)d_in[3];
  const float* W2    = (const float*)d_in[4];
  const float* b2    = (const float*)d_in[5];
  const float* gamma = (const float*)d_in[6];
  const float* beta  = (const float*)d_in[7];

  const int n_nodes = in_sizes[0] / DIM;
  const int n_edges = in_sizes[1] / 2;
  const int* srcI = edge;
  const int* dstI = edge + n_edges;

  const size_t NM = (size_t)n_nodes * DIM;
  float* B0    = (float*)d_ws;
  float* B1    = B0 + NM;
  float* stats = B1 + NM;     // 256 floats (sum, sumsq)
  float* ss    = stats + 256; // 256 floats (scale, shift)

  const int n4          = (int)(NM / 4);
  const int copy_blk    = (n4 + 255) / 256;
  const int scat_thr    = n_edges * 32;
  const int scat_blk    = (scat_thr + 255) / 256;
  const int mtiles      = (n_nodes + 15) / 16;
  const int gemm_blk    = (mtiles + 7) / 8;
  const size_t gemm_lds = (size_t)DIM * LDS_STRIDE * sizeof(float);
  const int apply_blk   = (int)((NM + 255) / 256);
  const float inv_n     = 1.0f / (float)n_nodes;

  const float* cur = x;
  float* aggs[3] = {B0, B1, B0};
  float* mids[3] = {B1, B0, B1};
  float* outs[3] = {B0, B1, (float*)d_out};

  for (int l = 0; l < 3; ++l) {
    float* agg = aggs[l];
    float* mid = mids[l];
    float* op  = outs[l];

    // agg = (1+eps)*x + sum_{e: dst=v} x[src]   (eps = 0)
    copy_f4_kernel<<<copy_blk, 256, 0, stream>>>((const float4*)cur, (float4*)agg, n4);
    scatter_add_kernel<<<scat_blk, 256, 0, stream>>>(cur, srcI, dstI, agg, n_edges);

    // MLP: LeakyReLU(agg @ W1^T + b1) @ W2^T + b2
    gemm_kernel<true><<<gemm_blk, 256, gemm_lds, stream>>>(agg, W1 + l * DIM * DIM,
                                                           b1 + l * DIM, mid, n_nodes);
    gemm_kernel<false><<<gemm_blk, 256, gemm_lds, stream>>>(mid, W2 + l * DIM * DIM,
                                                            b2 + l * DIM, op, n_nodes);

    // BatchNorm (training mode, batch stats)
    bn_zero_kernel<<<1, 256, 0, stream>>>(stats);
    bn_stats_kernel<<<512, 128, 0, stream>>>(op, stats, n_nodes);
    bn_finalize_kernel<<<1, 128, 0, stream>>>(stats, gamma + l * DIM, beta + l * DIM, ss, inv_n);
    bn_apply_kernel<<<apply_blk, 256, 0, stream>>>(op, ss, (int)NM);

    cur = op;
  }
}